// BCSRGraphConv_29918742184458
// MI455X (gfx1250) — compile-verified
//
#include <hip/hip_runtime.h>

typedef __attribute__((ext_vector_type(2))) float v2f;
typedef __attribute__((ext_vector_type(8))) float v8f;

#define N_NODES 100000
#define N_EDGES 1600000
#define D       128
#define AGG_CHUNK 256   // divides N_EDGES exactly; 8 batches of 32 per wave

// ---------------------------------------------------------------------------
// Kernel 1: fused dual GEMM via V_WMMA_F32_16X16X4_F32 (wave32).
//   XWn = X @ Wn   (into workspace, consumed by aggregation kernel)
//   OUT = X @ Ws   (self branch; initializes d_out)
// Each wave: 16 rows x 64 cols, both weight matrices, shared A fragments.
// ---------------------------------------------------------------------------
__global__ __launch_bounds__(256) void bcsr_dual_gemm_kernel(
    const float* __restrict__ X,
    const float* __restrict__ Wn,
    const float* __restrict__ Ws,
    float* __restrict__ XWn,
    float* __restrict__ OUT)
{
    const int lane = threadIdx.x & 31;
    const int wave = threadIdx.x >> 5;           // 0..7 (8 waves / block)
    const int rb   = blockIdx.x * 4 + (wave >> 1);
    const int row0 = rb * 16;
    if (row0 >= N_NODES) return;                 // wave-uniform -> EXEC all 1s
    const int col0 = (wave & 1) * 64;

    const int m    = lane & 15;                  // row within tile (A), col (B/D)
    const int koff = (lane >> 4) * 2;            // K-pair select per half-wave

    v8f accN[4], accS[4];
    const v8f vzero = {0.f,0.f,0.f,0.f,0.f,0.f,0.f,0.f};
#pragma unroll
    for (int t = 0; t < 4; ++t) { accN[t] = vzero; accS[t] = vzero; }

    const float* xrow = X + (size_t)(row0 + m) * D;

#pragma unroll 2
    for (int k = 0; k < D; k += 4) {
        const int kr = k + koff;
        // A fragment, 16x4 f32: lanes 0-15 K={k,k+1}, lanes 16-31 K={k+2,k+3}
        v2f a;
        a.x = xrow[kr];
        a.y = xrow[kr + 1];
#pragma unroll
        for (int t = 0; t < 4; ++t) {
            const int col = col0 + t * 16 + m;
            // B fragment, 4x16 f32: per-lane column, K pair matching A's halves
            v2f bn, bs;
            bn.x = Wn[(size_t)kr * D + col];
            bn.y = Wn[(size_t)(kr + 1) * D + col];
            bs.x = Ws[(size_t)kr * D + col];
            bs.y = Ws[(size_t)(kr + 1) * D + col];
            accN[t] = __builtin_amdgcn_wmma_f32_16x16x4_f32(
                false, a, false, bn, (short)0, accN[t], false, false);
            accS[t] = __builtin_amdgcn_wmma_f32_16x16x4_f32(
                false, a, false, bs, (short)0, accS[t], false, false);
        }
    }

    // D layout 16x16 f32: VGPR v -> row v (lanes 0-15) / row v+8 (lanes 16-31)
    const int rsel = (lane >> 4) * 8;
#pragma unroll
    for (int t = 0; t < 4; ++t) {
        const int col = col0 + t * 16 + m;
#pragma unroll
        for (int v = 0; v < 8; ++v) {
            const size_t idx = (size_t)(row0 + v + rsel) * D + col;
            XWn[idx] = accN[t][v];
            OUT[idx] = accS[t][v];
        }
    }
}

// ---------------------------------------------------------------------------
// Kernel 2: edge aggregation. edge_row is sorted, so each wave walks a
// contiguous chunk of 256 edges, run-length accumulating per destination row
// in registers (lane l owns features 4l..4l+3) and flushing with hardware
// f32 atomics only on row transitions / chunk boundaries.
//
// Edge metadata is staged lane-parallel: 3 coalesced loads per 32 edges,
// then broadcast with v_readlane (unrolled __shfl), so the steady-state loop
// issues only the 512B/edge coalesced xw gather as vmem traffic.
// ---------------------------------------------------------------------------
__device__ __forceinline__ void flush_row(float* __restrict__ dst,
                                          float ax, float ay, float az, float aw)
{
    __hip_atomic_fetch_add(dst + 0, ax, __ATOMIC_RELAXED, __HIP_MEMORY_SCOPE_AGENT);
    __hip_atomic_fetch_add(dst + 1, ay, __ATOMIC_RELAXED, __HIP_MEMORY_SCOPE_AGENT);
    __hip_atomic_fetch_add(dst + 2, az, __ATOMIC_RELAXED, __HIP_MEMORY_SCOPE_AGENT);
    __hip_atomic_fetch_add(dst + 3, aw, __ATOMIC_RELAXED, __HIP_MEMORY_SCOPE_AGENT);
}

__global__ __launch_bounds__(256) void bcsr_edge_agg_kernel(
    const float* __restrict__ XW,
    const long long* __restrict__ erow,
    const long long* __restrict__ ecol,
    const float* __restrict__ evals,
    float* __restrict__ OUT)
{
    const int lane    = threadIdx.x & 31;
    const int wave_id = (int)((blockIdx.x * blockDim.x + threadIdx.x) >> 5);
    const long long e0 = (long long)wave_id * AGG_CHUNK;
    if (e0 >= N_EDGES) return;                   // full chunks only (256 | E)

    const int f = lane * 4;                      // 4 features per lane, 128 total
    float ax = 0.f, ay = 0.f, az = 0.f, aw = 0.f;
    int cur = (int)erow[e0];                     // first destination row

    for (int b = 0; b < AGG_CHUNK; b += 32) {
        const long long eb = e0 + b;
        // Coalesced lane-parallel metadata stage: lane l owns edge eb+l
        const int   rowl = (int)erow[eb + lane];
        const int   coll = (int)ecol[eb + lane];
        const float vall = evals[eb + lane];
#pragma unroll
        for (int i = 0; i < 32; ++i) {
            const int   r = __shfl(rowl, i, 32); // v_readlane, imm lane
            const int   c = __shfl(coll, i, 32);
            const float v = __shfl(vall, i, 32);
            if (r != cur) {                      // uniform-valued branch
                flush_row(OUT + (size_t)cur * D + f, ax, ay, az, aw);
                ax = ay = az = aw = 0.f;
                cur = r;
            }
            const float4 x = *(const float4*)(XW + (size_t)c * D + f); // b128
            ax += x.x * v;
            ay += x.y * v;
            az += x.z * v;
            aw += x.w * v;
        }
    }
    flush_row(OUT + (size_t)cur * D + f, ax, ay, az, aw);
}

// ---------------------------------------------------------------------------
extern "C" void kernel_launch(void* const* d_in, const int* in_sizes, int n_in,
                              void* d_out, int out_size, void* d_ws, size_t ws_size,
                              hipStream_t stream)
{
    (void)in_sizes; (void)n_in; (void)out_size; (void)ws_size;
    const float*     X     = (const float*)d_in[0];
    const long long* erow  = (const long long*)d_in[1];
    const long long* ecol  = (const long long*)d_in[2];
    const float*     evals = (const float*)d_in[3];
    const float*     Wn    = (const float*)d_in[4];
    const float*     Ws    = (const float*)d_in[5];
    float* out = (float*)d_out;
    float* xw  = (float*)d_ws;                   // N_NODES * D * 4B = 51.2 MB

    // GEMM: 6250 row-blocks, 4 row-blocks x 2 col-halves per 256-thread block
    const int rb_total = (N_NODES + 15) / 16;    // 6250
    const int blocks1  = (rb_total + 3) / 4;     // 1563
    bcsr_dual_gemm_kernel<<<blocks1, 256, 0, stream>>>(X, Wn, Ws, xw, out);

    // Aggregation: one wave per 256-edge chunk
    const int waves2  = N_EDGES / AGG_CHUNK;                    // 6250
    const int blocks2 = (waves2 + 7) / 8;                       // 782
    bcsr_edge_agg_kernel<<<blocks2, 256, 0, stream>>>(xw, erow, ecol, evals, out);
}